// ExpandEvecs_75780402970966
// MI455X (gfx1250) — compile-verified
//
#include <hip/hip_runtime.h>
#include <hip/hip_bf16.h>

typedef __attribute__((ext_vector_type(2))) float v2f;
typedef __attribute__((ext_vector_type(4))) float v4f;
typedef __attribute__((ext_vector_type(8))) float v8f;

// out[b,k,n,m] = sum_{j<=k} e[b,n,j]*e[b,m,j]
// B=4, C=1, N=1024, K=16. One wave32 computes one 16x16 (n,m) tile for all
// 16 levels via a chain of V_WMMA_F32_16X16X4_F32 rank-1 updates (one-hot K).
__global__ __launch_bounds__(256)
void ExpandEvecs_75780402970966_kernel(const float* __restrict__ evecs,
                                       float* __restrict__ out) {
    constexpr int N  = 1024;
    constexpr int K  = 16;
    constexpr int NT = N / 16;           // 64 tiles per dim
    constexpr int TILES_PER_B = NT * NT; // 4096

    const int tid    = blockIdx.x * blockDim.x + threadIdx.x;
    const int waveId = tid >> 5;                 // global wave index
    const int lane   = threadIdx.x & 31;
    const int r      = lane & 15;
    const bool lo    = lane < 16;

    const int b   = waveId / TILES_PER_B;
    const int rem = waveId % TILES_PER_B;
    const int n0  = (rem / NT) * 16;
    const int m0  = (rem % NT) * 16;

    // Each lane loads full 16-float rows for its n-tile row and m-tile row.
    // Input is 256 KB total -> L2-resident; duplicate loads across half-waves
    // are free. Rows are 64 B aligned -> four b128 loads each.
    const float* rowN = evecs + (((size_t)b * N + (n0 + r)) * K);
    const float* rowM = evecs + (((size_t)b * N + (m0 + r)) * K);

    float an[16], bm[16];
    {
        v4f t0 = *(const v4f*)(rowN + 0);
        v4f t1 = *(const v4f*)(rowN + 4);
        v4f t2 = *(const v4f*)(rowN + 8);
        v4f t3 = *(const v4f*)(rowN + 12);
        *(v4f*)&an[0] = t0; *(v4f*)&an[4] = t1;
        *(v4f*)&an[8] = t2; *(v4f*)&an[12] = t3;
        v4f u0 = *(const v4f*)(rowM + 0);
        v4f u1 = *(const v4f*)(rowM + 4);
        v4f u2 = *(const v4f*)(rowM + 8);
        v4f u3 = *(const v4f*)(rowM + 12);
        *(v4f*)&bm[0] = u0; *(v4f*)&bm[4] = u1;
        *(v4f*)&bm[8] = u2; *(v4f*)&bm[12] = u3;
    }

    v8f acc = {};  // running low-rank accumulator (16x16 f32, 8 VGPRs)

    // D-tile store geometry: VGPR v, lane l -> row n0 + v + (l>=16 ? 8 : 0),
    // col m0 + (l&15). Row segments of 64 B; adjacent waves tile full rows.
    const int rowBase = n0 + (lo ? 0 : 8);
    float* dstCol = out + (size_t)(b * K) * ((size_t)N * N) + (size_t)(m0 + r);

#pragma unroll
    for (int k = 0; k < K; ++k) {
        // A (16x4): only column K=0 nonzero = e[n-tile, k].
        // VGPR0 lanes 0-15 = K=0, lanes 16-31 = K=2 (zeroed); VGPR1 = K=1,3 (zero).
        v2f av; av.x = lo ? an[k] : 0.0f; av.y = 0.0f;
        // B (4x16): only row K=0 nonzero = e[m-tile, k].
        v2f bv; bv.x = lo ? bm[k] : 0.0f; bv.y = 0.0f;

        // acc = e_k e_k^T + acc   (exactly the cumsum recurrence)
        acc = __builtin_amdgcn_wmma_f32_16x16x4_f32(
            /*neg_a=*/false, av, /*neg_b=*/false, bv,
            /*c_mod=*/(short)0, acc, /*reuse_a=*/false, /*reuse_b=*/false);

        // Stream level k to memory. Output (256 MB) exceeds L2 (192 MB) and is
        // never re-read -> non-temporal stores.
        float* dstLvl = dstCol + (size_t)k * ((size_t)N * N);
#pragma unroll
        for (int v = 0; v < 8; ++v) {
            __builtin_nontemporal_store(acc[v],
                dstLvl + (size_t)(rowBase + v) * N);
        }
    }
}

extern "C" void kernel_launch(void* const* d_in, const int* in_sizes, int n_in,
                              void* d_out, int out_size, void* d_ws, size_t ws_size,
                              hipStream_t stream) {
    const float* evecs = (const float*)d_in[0];
    float* out = (float*)d_out;

    // 4 batches * 64*64 tiles = 16384 waves; 8 waves (256 threads) per block.
    constexpr int totalWaves = 4 * 64 * 64;
    constexpr int blockSize  = 256;
    constexpr int wavesPerBlock = blockSize / 32;
    constexpr int grid = totalWaves / wavesPerBlock;  // 2048

    ExpandEvecs_75780402970966_kernel<<<grid, blockSize, 0, stream>>>(evecs, out);
}